// CapsuleCONV_1709396984016
// MI455X (gfx1250) — compile-verified
//
#include <hip/hip_runtime.h>

typedef __attribute__((ext_vector_type(2))) float v2f;
typedef __attribute__((ext_vector_type(4))) float v4f;
typedef __attribute__((ext_vector_type(8))) float v8f;

// xor-shuffle within 16-lane halves via ds_swizzle (group-of-32 mode,
// and_mask=0x1f, or_mask=0, xor_mask=MASK). MASK<16 keeps halves separate.
template<int MASK>
__device__ __forceinline__ float swz_xor(float v) {
    int i = __builtin_amdgcn_ds_swizzle(__builtin_bit_cast(int, v), 0x7C00 | MASK);
    return __builtin_bit_cast(float, i);
}
__device__ __forceinline__ float half_max(float v) {
    v = fmaxf(v, swz_xor<1>(v));
    v = fmaxf(v, swz_xor<2>(v));
    v = fmaxf(v, swz_xor<4>(v));
    v = fmaxf(v, swz_xor<8>(v));
    return v;
}
__device__ __forceinline__ float half_sum(float v) {
    v += swz_xor<1>(v);
    v += swz_xor<2>(v);
    v += swz_xor<4>(v);
    v += swz_xor<8>(v);
    return v;
}

// input (32,32,32,32,16); ncv (32,32,15,15,16); w (3,3,32,4,4,32)
// out (32,32,15,15,16). One wave handles 4 consecutive flat sites (b*225+h*15+w).
__global__ __launch_bounds__(32) void caps_attn_kernel(
    const float* __restrict__ inp, const float* __restrict__ ncv,
    const float* __restrict__ w, const float* __restrict__ gamma,
    const float* __restrict__ beta, float* __restrict__ outp)
{
    const int lane = threadIdx.x;          // 0..31
    const int c    = lane & 15;            // column / m-low
    const int hi   = lane >> 4;            // lane half
    const int x0   = hi << 1;              // K (x) pair base for A and B
    const int quad = blockIdx.x;           // 0..1799

    // ---- A addressing: this lane supplies A row r=c -> site quad*4+(c>>2), a=c&3
    {
    }
    const int sA = quad * 4 + (c >> 2);
    const int aA = c & 3;
    const int bA = sA / 225;
    const int rA = sA % 225;
    const int hA = rA / 15;
    const int wAx = rA % 15;
    const int pbaseA = bA * 524288 + (2 * hA) * 512 + (2 * wAx) * 16 + aA * 4 + x0;

    // ---- D/ncv/out ownership: this lane owns (s = quad*4 + 2*hi + b, m = c + 16*p)
    int nb[4];
#pragma unroll
    for (int b = 0; b < 2; b++) {
        const int sD = quad * 4 + 2 * hi + b;
        const int bD = sD / 225;
        const int rD = sD % 225;
        const int hD = rD / 15;
        const int wD = rD % 15;
#pragma unroll
        for (int p = 0; p < 2; p++) {
            const int m = c + 16 * p;
            nb[b * 2 + p] = ((bD * 32 + m) * 225 + hD * 15 + wD) * 16;
        }
    }

    // ---- preload ncv[s][m][a*4+d] (16 contiguous floats per owned (s,m) pair)
    v4f nr[4][4];
#pragma unroll
    for (int pr = 0; pr < 4; pr++)
#pragma unroll
        for (int a = 0; a < 4; a++)
            nr[pr][a] = *(const v4f*)(ncv + nb[pr] + a * 4);

    const int wlane = x0 * 128 + c;        // B element offset: x0*128 + col

    v8f acc[8];
#pragma unroll
    for (int t = 0; t < 8; t++) acc[t] = (v8f){};

#pragma unroll 1
    for (int n = 0; n < 32; n++) {
#pragma unroll 1
        for (int kk = 0; kk < 3; kk++) {
#pragma unroll 1
            for (int ll = 0; ll < 3; ll++) {
                // A: inp[s, n, 2h+kk, 2w+ll, aA*4 + {x0,x0+1}]
                const float* ap = inp + pbaseA + n * 16384 + kk * 512 + ll * 16;
                const v2f A = *(const v2f*)ap;
                // B base for w[kk,ll,n,:,:,:] (x*128 + d*32 + m layout)
                const float* wp = w + ((kk * 3 + ll) * 32 + n) * 512 + wlane;

                // votes: V[t] = A(16x4) x B(4x16) over chunk t of (d,m)
                v8f V[8];
#pragma unroll
                for (int t = 0; t < 8; t++) {
                    v2f Bv;
                    Bv.x = wp[t * 16];
                    Bv.y = wp[t * 16 + 128];
                    v8f z = (v8f){};
                    V[t] = __builtin_amdgcn_wmma_f32_16x16x4_f32(
                        false, A, false, Bv, (short)0, z, false, false);
                }

                // qk[s,m] = sum_{a,d} V[(s,a),(d,m)] * ncv[s,m,a,d] -- fully in-lane
                float qv[2][2] = {{0.f, 0.f}, {0.f, 0.f}};
#pragma unroll
                for (int b = 0; b < 2; b++)
#pragma unroll
                    for (int p = 0; p < 2; p++)
#pragma unroll
                        for (int a = 0; a < 4; a++) {
                            const v4f nc = nr[b * 2 + p][a];
                            const int j = 4 * b + a;
                            qv[b][p] = fmaf(V[0 + p][j], nc.x, qv[b][p]);
                            qv[b][p] = fmaf(V[2 + p][j], nc.y, qv[b][p]);
                            qv[b][p] = fmaf(V[4 + p][j], nc.z, qv[b][p]);
                            qv[b][p] = fmaf(V[6 + p][j], nc.w, qv[b][p]);
                        }

                // softmax over m (16 lanes in half x 2 p-slots), then renorm
                float q[2][2];
#pragma unroll
                for (int b = 0; b < 2; b++) {
                    const float q0 = qv[b][0] * 0.25f;   // scale = 1/sqrt(16)
                    const float q1 = qv[b][1] * 0.25f;
                    const float mx = half_max(fmaxf(q0, q1));
                    const float e0 = expf(q0 - mx);
                    const float e1 = expf(q1 - mx);
                    const float se = half_sum(e0 + e1);
                    const float p0 = e0 / se;
                    const float p1 = e1 / se;
                    const float s2 = half_sum(p0 + p1);  // == sum(softmax)
                    const float inv = 1.0f / (s2 + 1e-10f);
                    q[b][0] = p0 * inv;
                    q[b][1] = p1 * inv;
                }

                // out[s,m,a,d] += q[s,m] * V  (same lane layout as V)
#pragma unroll
                for (int t = 0; t < 8; t++) {
                    const int p = t & 1;
#pragma unroll
                    for (int j = 0; j < 8; j++)
                        acc[t][j] = fmaf(q[j >> 2][p], V[t][j], acc[t][j]);
                }
            }
        }
    }

    // ---- LayerNorm over the 16 (a,d) values of each owned (s,m), then store
    float g[16], bb[16];
#pragma unroll
    for (int i = 0; i < 16; i++) { g[i] = gamma[i]; bb[i] = beta[i]; }

#pragma unroll
    for (int b = 0; b < 2; b++)
#pragma unroll
        for (int p = 0; p < 2; p++) {
            float s = 0.f, ss = 0.f;
#pragma unroll
            for (int a = 0; a < 4; a++)
#pragma unroll
                for (int d = 0; d < 4; d++) {
                    const float v = acc[2 * d + p][4 * b + a];
                    s += v;
                    ss += v * v;
                }
            const float mu  = s * (1.0f / 16.0f);
            const float var = ss * (1.0f / 16.0f) - mu * mu;
            const float rstd = 1.0f / sqrtf(var + 1e-5f);
#pragma unroll
            for (int a = 0; a < 4; a++) {
                v4f o;
#pragma unroll
                for (int d = 0; d < 4; d++) {
                    const int dd = a * 4 + d;
                    o[d] = (acc[2 * d + p][4 * b + a] - mu) * rstd * g[dd] + bb[dd];
                }
                *(v4f*)(outp + nb[b * 2 + p] + a * 4) = o;
            }
        }
}

extern "C" void kernel_launch(void* const* d_in, const int* in_sizes, int n_in,
                              void* d_out, int out_size, void* d_ws, size_t ws_size,
                              hipStream_t stream) {
    (void)in_sizes; (void)n_in; (void)out_size; (void)d_ws; (void)ws_size;
    const float* inp   = (const float*)d_in[0];
    const float* ncv   = (const float*)d_in[1];
    const float* w     = (const float*)d_in[2];
    const float* gamma = (const float*)d_in[3];
    const float* beta  = (const float*)d_in[4];
    float* out = (float*)d_out;
    // 7200 sites / 4 per wave = 1800 single-wave workgroups
    caps_attn_kernel<<<dim3(1800), dim3(32), 0, stream>>>(inp, ncv, w, gamma, beta, out);
}